// RNNModel_51118700757092
// MI455X (gfx1250) — compile-verified
//
#include <hip/hip_runtime.h>
#include <math.h>

typedef float v2f __attribute__((ext_vector_type(2)));
typedef float v8f __attribute__((ext_vector_type(8)));

#define VOCAB   57
#define IN_F    30
#define IN_P    32          // emb row padded to 32 floats in LDS (zero pad)
#define HID     40
#define OUT_F   18
#define T_STEPS 100
#define BATCH   8192

#define HID_P   48          // HID padded to 3 N-tiles of 16
#define KC_H    10          // K-chunks of 4 covering HID=40 exactly
#define KC_X    8           // K-chunks of 4 covering IN=30 (padded to 32)
#define NT      3           // N tiles (48/16)

#define WPB            4    // waves per block
#define ROWS_PER_WAVE  16
#define BLOCK_THREADS  (WPB * 32)
#define ROWS_PER_BLOCK (WPB * ROWS_PER_WAVE)   // 64

// fast tanh: one v_exp_f32 + a few VALU, co-executes with the WMMA pipe
__device__ __forceinline__ float fast_tanh(float x) {
    float t = __expf(2.0f * x);
    return (t - 1.0f) / (t + 1.0f);
}

__global__ __launch_bounds__(BLOCK_THREADS)
void rnn_fused_wmma(const int*   __restrict__ input_idx,
                    const float* __restrict__ hidden0,
                    const float* __restrict__ emb,
                    const float* __restrict__ W_ih,
                    const float* __restrict__ W_hh,
                    const float* __restrict__ b_ih,
                    const float* __restrict__ b_hh,
                    const float* __restrict__ W_fc,
                    const float* __restrict__ b_fc,
                    float*       __restrict__ out)
{
    __shared__ float s_emb[VOCAB * IN_P];                       // 7.3 KB, rows zero-padded
    __shared__ float s_h[WPB][ROWS_PER_WAVE][HID_P];            // 12.0 KB

    const int tid   = threadIdx.x;
    const int wave  = tid >> 5;
    const int lane  = tid & 31;
    const int m     = lane & 15;                 // tile row (batch) this lane maps to
    const int khalf = (lane >> 4) << 1;          // f32 A/B frag: lanes 16-31 hold K+2,K+3
    const int kslot = khalf >> 1;                // v2f slot offset (0 or 1)
    const int blockBase = blockIdx.x * ROWS_PER_BLOCK;
    const int grow  = blockBase + wave * ROWS_PER_WAVE + m;   // global batch row

    // ---- cooperative setup: padded embedding table + initial hidden into LDS ----
    for (int i = tid; i < VOCAB * IN_P; i += BLOCK_THREADS) {
        int col = i & (IN_P - 1);
        s_emb[i] = (col < IN_F) ? emb[(i >> 5) * IN_F + col] : 0.f;
    }
    for (int i = tid; i < WPB * ROWS_PER_WAVE * HID_P; i += BLOCK_THREADS) {
        int w = i / (ROWS_PER_WAVE * HID_P);
        int r = (i / HID_P) % ROWS_PER_WAVE;
        int j = i % HID_P;
        float v = 0.f;
        if (j < HID)
            v = hidden0[(size_t)(blockBase + w * ROWS_PER_WAVE + r) * HID + j];
        s_h[w][r][j] = v;
    }
    __syncthreads();

    // ---- per-lane constant B-fragments (kept in VGPRs for all 100 steps) ----
    // B[k][n] = W^T, n = nt*16 + (lane&15); V0 holds K=kbase+khalf, V1 pairs via .y
    v2f   bWhh[NT][KC_H];
    v2f   bWih[NT][KC_X];
    float biasN[NT];
    for (int nt = 0; nt < NT; ++nt) {
        const int  n   = nt * 16 + m;
        const bool nok = (n < HID);
        biasN[nt] = nok ? (b_ih[n] + b_hh[n]) : 0.f;
        for (int kc = 0; kc < KC_H; ++kc) {
            const int k0 = kc * 4 + khalf;
            v2f v; v.x = 0.f; v.y = 0.f;
            if (nok) { v.x = W_hh[n * HID + k0]; v.y = W_hh[n * HID + k0 + 1]; }
            bWhh[nt][kc] = v;
        }
        for (int kc = 0; kc < KC_X; ++kc) {
            const int k0 = kc * 4 + khalf;
            v2f v; v.x = 0.f; v.y = 0.f;
            if (nok) {
                if (k0     < IN_F) v.x = W_ih[n * IN_F + k0];
                if (k0 + 1 < IN_F) v.y = W_ih[n * IN_F + k0 + 1];
            }
            bWih[nt][kc] = v;
        }
    }

    // ---- the 100-step recurrence: h = tanh(x@W_ih^T + b + h@W_hh^T) ----
    for (int t = 0; t < T_STEPS; ++t) {
        // embedding index for this lane's batch row (1 dword / lane / step)
        const int idx = input_idx[(size_t)t * BATCH + grow];

        // A-fragments for x: unconditional v2f LDS loads from the padded table
        const v2f* erow2 = (const v2f*)&s_emb[idx * IN_P];
        v2f aX[KC_X];
        for (int kc = 0; kc < KC_X; ++kc)
            aX[kc] = erow2[kc * 2 + kslot];

        // A-fragments for h (LDS round-trip performs the D->A transpose)
        const v2f* hrow2 = (const v2f*)&s_h[wave][m][0];
        v2f aH[KC_H];
        for (int kc = 0; kc < KC_H; ++kc)
            aH[kc] = hrow2[kc * 2 + kslot];

        // seed accumulators with (b_ih + b_hh): C layout has N = lane&15 in all 8 regs
        v8f acc[NT];
        for (int nt = 0; nt < NT; ++nt) {
            v8f c;
            for (int r = 0; r < 8; ++r) c[r] = biasN[nt];
            acc[nt] = c;
        }

        // 54 fp32 WMMAs: input projection (8 K-chunks) + recurrence (10 K-chunks)
        for (int kc = 0; kc < KC_X; ++kc)
            for (int nt = 0; nt < NT; ++nt)
                acc[nt] = __builtin_amdgcn_wmma_f32_16x16x4_f32(
                    false, aX[kc], false, bWih[nt][kc], (short)0, acc[nt], false, false);
        for (int kc = 0; kc < KC_H; ++kc)
            for (int nt = 0; nt < NT; ++nt)
                acc[nt] = __builtin_amdgcn_wmma_f32_16x16x4_f32(
                    false, aH[kc], false, bWhh[nt][kc], (short)0, acc[nt], false, false);

        // tanh + scatter D back to LDS h buffer (D layout: M = r + 8*(lane>=16))
        const int mhi = (lane >> 4) << 3;
        for (int nt = 0; nt < NT; ++nt) {
            const int n = nt * 16 + m;
            for (int r = 0; r < 8; ++r)
                s_h[wave][r + mhi][n] = fast_tanh(acc[nt][r]);
        }
        __syncthreads();   // order step-t stores vs step-(t+1) cross-lane loads
    }

    // ---- final FC + sigmoid: tiny (16x18x40), lanes 0-15 handle their rows ----
    if (lane < 16) {
        for (int o = 0; o < OUT_F; ++o) {
            float s = b_fc[o];
            for (int j = 0; j < HID; ++j)
                s += s_h[wave][m][j] * W_fc[o * HID + j];
            out[(size_t)grow * OUT_F + o] = 1.f / (1.f + __expf(-s));
        }
    }
}

extern "C" void kernel_launch(void* const* d_in, const int* in_sizes, int n_in,
                              void* d_out, int out_size, void* d_ws, size_t ws_size,
                              hipStream_t stream) {
    const int*   input_idx = (const int*)  d_in[0];
    const float* hidden0   = (const float*)d_in[1];
    const float* emb       = (const float*)d_in[2];
    const float* W_ih      = (const float*)d_in[3];
    const float* W_hh      = (const float*)d_in[4];
    const float* b_ih      = (const float*)d_in[5];
    const float* b_hh      = (const float*)d_in[6];
    const float* W_fc      = (const float*)d_in[7];
    const float* b_fc      = (const float*)d_in[8];
    float* out = (float*)d_out;

    const int blocks = BATCH / ROWS_PER_BLOCK;   // 128
    rnn_fused_wmma<<<blocks, BLOCK_THREADS, 0, stream>>>(
        input_idx, hidden0, emb, W_ih, W_hh, b_ih, b_hh, W_fc, b_fc, out);
}